// ChebyASPIRELayer_49555332662210
// MI455X (gfx1250) — compile-verified
//
#include <hip/hip_runtime.h>
#include <cstdint>
#include <cstddef>

// ===========================================================================
// ChebyASPIRE on MI455X (gfx1250).
// Memory-bound sparse filter: S(v) = (X^T (X v) - t_mid v)/t_half applied 20x.
// Strategy: build CSR(users) + CSC(items) once per launch -> atomic-free,
// gather-only row-parallel SpMM (1 wave32 per 128-wide row, float4/lane),
// with the (idx,val) stream double-buffered into LDS via CDNA5 async-to-LDS
// loads (ASYNCcnt in-order completion lets the prefetch stay in flight).
// Working set (~100 MB) is L2-resident (192 MB), so repeated gathers hit L2.
// ===========================================================================

#define BATCHN 128
#define NUSERS 50000

typedef __attribute__((address_space(1))) int g1_int;  // global (AS1) int
typedef __attribute__((address_space(3))) int l3_int;  // LDS    (AS3) int

__device__ __forceinline__ void cdna5_async_ld_b32(void* lds, const void* g) {
#if __has_builtin(__builtin_amdgcn_global_load_async_to_lds_b32)
  __builtin_amdgcn_global_load_async_to_lds_b32(
      (g1_int*)g, (l3_int*)lds, /*offset=*/0, /*cpol=*/0);
#else
  *(volatile int*)lds = *(const volatile int*)g;  // functional fallback
#endif
}

__device__ __forceinline__ void cdna5_wait_async_le0() {
#if __has_builtin(__builtin_amdgcn_s_wait_asynccnt)
  __builtin_amdgcn_s_wait_asynccnt(0);
#else
  asm volatile("s_wait_asynccnt 0x0" ::: "memory");
#endif
  asm volatile("" ::: "memory");  // compiler barrier: LDS reads stay after wait
}

// Allow the newest 2 async loads (the prefetched next chunk) to remain in
// flight; async loads complete in order, so older ones are guaranteed done.
__device__ __forceinline__ void cdna5_wait_async_le2() {
#if __has_builtin(__builtin_amdgcn_s_wait_asynccnt)
  __builtin_amdgcn_s_wait_asynccnt(2);
#else
  asm volatile("s_wait_asynccnt 0x2" ::: "memory");
#endif
  asm volatile("" ::: "memory");
}

// ---------------------------------------------------------------------------
// Row-parallel SpMM: dst[row,0:128] = sum_e val[e] * src[idx[e],0:128]
// One wave32 per row; lane j owns elements [4j..4j+3] (float4).
// (idx,val) stream staged through 2 LDS buffers per wave with async-to-LDS.
// ---------------------------------------------------------------------------
#define SPMM_BLOCK 256
#define WPB (SPMM_BLOCK / 32)

__global__ __launch_bounds__(SPMM_BLOCK)
void spmm_csr_kernel(float* __restrict__ dst, const float* __restrict__ src,
                     const int* __restrict__ offs, const int* __restrict__ idx,
                     const float* __restrict__ val, int nrows) {
  __shared__ int   sIdx[WPB][2][32];
  __shared__ float sVal[WPB][2][32];
  const int lane = threadIdx.x & 31;
  const int wave = threadIdx.x >> 5;
  const int row  = blockIdx.x * WPB + wave;
  if (row >= nrows) return;
  const int start = offs[row];
  const int end   = offs[row + 1];
  float a0 = 0.f, a1 = 0.f, a2 = 0.f, a3 = 0.f;

  // Prefetch first chunk into buffer 0.
  if (start < end) {
    const int m0 = (end - start < 32) ? (end - start) : 32;
    if (lane < m0) {
      cdna5_async_ld_b32(&sIdx[wave][0][lane], idx + start + lane);
      cdna5_async_ld_b32(&sVal[wave][0][lane], val + start + lane);
    }
  }

  int buf = 0;
  for (int p = start; p < end; p += 32, buf ^= 1) {
    const int m  = (end - p < 32) ? (end - p) : 32;
    const int pn = p + 32;
    if (pn < end) {
      // Prefetch next chunk into the other buffer, then wait only for the
      // current buffer (<=2 outstanding: the prefetch just issued).
      const int mn = (end - pn < 32) ? (end - pn) : 32;
      if (lane < mn) {
        cdna5_async_ld_b32(&sIdx[wave][buf ^ 1][lane], idx + pn + lane);
        cdna5_async_ld_b32(&sVal[wave][buf ^ 1][lane], val + pn + lane);
      }
      cdna5_wait_async_le2();
    } else {
      cdna5_wait_async_le0();
    }
    for (int t = 0; t < m; ++t) {
      const int   c = sIdx[wave][buf][t];   // uniform LDS broadcast
      const float v = sVal[wave][buf][t];
      const float4 s = *(const float4*)(src + (size_t)c * BATCHN + lane * 4);
      a0 = fmaf(v, s.x, a0);
      a1 = fmaf(v, s.y, a1);
      a2 = fmaf(v, s.z, a2);
      a3 = fmaf(v, s.w, a3);
    }
  }
  float4 o; o.x = a0; o.y = a1; o.z = a2; o.w = a3;
  *(float4*)(dst + (size_t)row * BATCHN + lane * 4) = o;
}

// ---------------------------------------------------------------------------
// small utility kernels
// ---------------------------------------------------------------------------
__global__ void zero_i32_kernel(int* p, int n) {
  int i = blockIdx.x * blockDim.x + threadIdx.x;
  if (i < n) p[i] = 0;
}

__global__ void copy_i32_kernel(int* dst, const int* __restrict__ src, int n) {
  int i = blockIdx.x * blockDim.x + threadIdx.x;
  if (i < n) dst[i] = src[i];
}

__global__ void count_rc_kernel(const int* __restrict__ rows,
                                const int* __restrict__ cols,
                                int* cntR, int* cntC, int nnz) {
  int e = blockIdx.x * blockDim.x + threadIdx.x;
  if (e < nnz) {
    atomicAdd(cntR + rows[e], 1);
    atomicAdd(cntC + cols[e], 1);
  }
}

// Single-block exclusive scan: cnt[0..n) -> offs[0..n] (offs[n] = total).
__global__ void scan_excl_kernel(const int* __restrict__ cnt,
                                 int* __restrict__ offs, int n) {
  __shared__ int s0[1024];
  __shared__ int s1[1024];
  const int T = 1024;
  const int t = threadIdx.x;
  const int chunk = (n + T - 1) / T;
  const int lo = t * chunk;
  const int hi = (lo + chunk < n) ? (lo + chunk) : n;
  int s = 0;
  for (int i = lo; i < hi; ++i) s += cnt[i];
  s0[t] = s;
  __syncthreads();
  int* src = s0;
  int* dst = s1;
  for (int off = 1; off < T; off <<= 1) {
    int v = src[t];
    if (t >= off) v += src[t - off];
    dst[t] = v;
    __syncthreads();
    int* tmp = src; src = dst; dst = tmp;
  }
  int run = (t == 0) ? 0 : src[t - 1];
  for (int i = lo; i < hi; ++i) { offs[i] = run; run += cnt[i]; }
  if (t == T - 1) offs[n] = src[T - 1];
}

__global__ void csr_fill_kernel(const int* __restrict__ rows,
                                const int* __restrict__ cols,
                                const float* __restrict__ vals,
                                int* curR, int* curC,
                                int* __restrict__ csrCol, float* __restrict__ csrVal,
                                int* __restrict__ cscRow, float* __restrict__ cscVal,
                                int nnz) {
  int e = blockIdx.x * blockDim.x + threadIdx.x;
  if (e >= nnz) return;
  const int r = rows[e];
  const int c = cols[e];
  const float v = vals[e];
  const int pr = atomicAdd(curR + r, 1);
  csrCol[pr] = c; csrVal[pr] = v;
  const int pc = atomicAdd(curC + c, 1);
  cscRow[pc] = r; cscVal[pc] = v;
}

// ---------------------------------------------------------------------------
// Chebyshev elementwise kernels (scalars read on-device: graph-capture safe)
// ---------------------------------------------------------------------------
__global__ void cheby_first_kernel(const float* __restrict__ T0,
                                   const float* __restrict__ W,
                                   float* __restrict__ T1,
                                   float* __restrict__ acc,
                                   const float* __restrict__ coeffs,
                                   const float* __restrict__ tmid,
                                   const float* __restrict__ thalf, int n) {
  int i = blockIdx.x * blockDim.x + threadIdx.x;
  if (i >= n) return;
  const float m    = tmid[0];
  const float invh = 1.0f / thalf[0];
  const float t0 = T0[i];
  const float t1 = (W[i] - m * t0) * invh;
  T1[i]  = t1;
  acc[i] = coeffs[0] * t0 + coeffs[1] * t1;
}

__global__ void cheby_step_kernel(float* __restrict__ Tp /* in: T_prev, out: T_next */,
                                  const float* __restrict__ Tc,
                                  const float* __restrict__ W,
                                  float* __restrict__ acc,
                                  const float* __restrict__ coeffs, int k,
                                  const float* __restrict__ tmid,
                                  const float* __restrict__ thalf, int n) {
  int i = blockIdx.x * blockDim.x + threadIdx.x;
  if (i >= n) return;
  const float m    = tmid[0];
  const float invh = 1.0f / thalf[0];
  const float ck   = coeffs[k];
  const float s  = (W[i] - m * Tc[i]) * invh;
  const float tn = 2.0f * s - Tp[i];
  Tp[i]   = tn;
  acc[i] += ck * tn;
}

// ---------------------------------------------------------------------------
// LDS-tiled transposes: X_batch [128 x ni] <-> [ni x 128]
// ---------------------------------------------------------------------------
__global__ void transpose_in_kernel(const float* __restrict__ X,
                                    float* __restrict__ A, int ni) {
  __shared__ float tile[32][33];
  const int i0 = blockIdx.x * 32;
  const int b0 = blockIdx.y * 32;
  const int tx = threadIdx.x, ty = threadIdx.y;
  for (int j = 0; j < 32; j += 8) {
    const int b = b0 + ty + j;
    const int i = i0 + tx;
    tile[ty + j][tx] = (i < ni) ? X[(size_t)b * ni + i] : 0.f;
  }
  __syncthreads();
  for (int j = 0; j < 32; j += 8) {
    const int i = i0 + ty + j;
    const int b = b0 + tx;
    if (i < ni) A[(size_t)i * BATCHN + b] = tile[tx][ty + j];
  }
}

__global__ void transpose_out_kernel(const float* __restrict__ ACC,
                                     float* __restrict__ out, int ni) {
  __shared__ float tile[32][33];
  const int i0 = blockIdx.x * 32;
  const int b0 = blockIdx.y * 32;
  const int tx = threadIdx.x, ty = threadIdx.y;
  for (int j = 0; j < 32; j += 8) {
    const int i = i0 + ty + j;
    tile[ty + j][tx] = (i < ni) ? ACC[(size_t)i * BATCHN + (b0 + tx)] : 0.f;
  }
  __syncthreads();
  for (int j = 0; j < 32; j += 8) {
    const int i = i0 + tx;
    const int b = b0 + ty + j;
    if (i < ni) out[(size_t)b * ni + i] = tile[tx][ty + j];
  }
}

// ---------------------------------------------------------------------------
// host orchestration (graph-capture safe: only kernel launches on `stream`)
// ---------------------------------------------------------------------------
extern "C" void kernel_launch(void* const* d_in, const int* in_sizes, int n_in,
                              void* d_out, int out_size, void* d_ws, size_t ws_size,
                              hipStream_t stream) {
  const float* X      = (const float*)d_in[0];
  const float* vals   = (const float*)d_in[1];
  const float* coeffs = (const float*)d_in[2];
  const float* tmid   = (const float*)d_in[3];
  const float* thalf  = (const float*)d_in[4];
  const int*   rows   = (const int*)d_in[5];
  const int*   cols   = (const int*)d_in[6];

  const int nnz = in_sizes[1];
  const int K1  = in_sizes[2];          // DEGREE + 1 coefficients
  const int ni  = in_sizes[0] / BATCHN; // n_items
  const int nu  = NUSERS;               // n_users (reference constant; device
                                        // scalar can't be read during capture)
  const int nE  = ni * BATCHN;          // elements per [ni x 128] matrix

  // --- carve workspace (256B aligned slices) ---
  char* wp = (char*)d_ws;
  auto take = [&](size_t bytes) {
    char* r = wp;
    wp += (bytes + 255) & ~(size_t)255;
    return (void*)r;
  };
  float* A   = (float*)take((size_t)nE * 4);        // T_prev
  float* Bm  = (float*)take((size_t)nE * 4);        // T_curr
  float* W   = (float*)take((size_t)nE * 4);        // X^T X v scratch
  float* ACC = (float*)take((size_t)nE * 4);        // accumulator
  float* U   = (float*)take((size_t)nu * BATCHN * 4); // user-space intermediate
  int*   rowOffs = (int*)take((size_t)(nu + 1) * 4);
  int*   colOffs = (int*)take((size_t)(ni + 1) * 4);
  int*   curR    = (int*)take((size_t)nu * 4);      // counts, then cursors
  int*   curC    = (int*)take((size_t)ni * 4);
  int*   csrCol  = (int*)take((size_t)nnz * 4);
  float* csrVal  = (float*)take((size_t)nnz * 4);
  int*   cscRow  = (int*)take((size_t)nnz * 4);
  float* cscVal  = (float*)take((size_t)nnz * 4);
  if ((size_t)(wp - (char*)d_ws) > ws_size) return;  // workspace too small

  const int TB = 256;
  const int gNnz = (nnz + TB - 1) / TB;
  const int gNu  = (nu + TB - 1) / TB;
  const int gNi  = (ni + TB - 1) / TB;
  const int gE   = (nE + TB - 1) / TB;

  // --- build CSR (users) + CSC (items), amortized over 40 SpMMs ---
  zero_i32_kernel<<<gNu, TB, 0, stream>>>(curR, nu);
  zero_i32_kernel<<<gNi, TB, 0, stream>>>(curC, ni);
  count_rc_kernel<<<gNnz, TB, 0, stream>>>(rows, cols, curR, curC, nnz);
  scan_excl_kernel<<<1, 1024, 0, stream>>>(curR, rowOffs, nu);
  scan_excl_kernel<<<1, 1024, 0, stream>>>(curC, colOffs, ni);
  copy_i32_kernel<<<gNu, TB, 0, stream>>>(curR, rowOffs, nu);
  copy_i32_kernel<<<gNi, TB, 0, stream>>>(curC, colOffs, ni);
  csr_fill_kernel<<<gNnz, TB, 0, stream>>>(rows, cols, vals, curR, curC,
                                           csrCol, csrVal, cscRow, cscVal, nnz);

  // --- T_prev = X_batch^T ---
  dim3 bT(32, 8);
  dim3 gT((ni + 31) / 32, BATCHN / 32);
  transpose_in_kernel<<<gT, bT, 0, stream>>>(X, A, ni);

  const int gSpU = (nu + WPB - 1) / WPB;
  const int gSpI = (ni + WPB - 1) / WPB;
  auto applyS = [&](const float* v) {
    // u = X v  (gather item rows, one wave per user row)
    spmm_csr_kernel<<<gSpU, SPMM_BLOCK, 0, stream>>>(U, v, rowOffs, csrCol, csrVal, nu);
    // w = X^T u (gather user rows, one wave per item row)
    spmm_csr_kernel<<<gSpI, SPMM_BLOCK, 0, stream>>>(W, U, colOffs, cscRow, cscVal, ni);
  };

  // --- T_curr = S(T_prev); acc = c0*T_prev + c1*T_curr ---
  applyS(A);
  cheby_first_kernel<<<gE, TB, 0, stream>>>(A, W, Bm, ACC, coeffs, tmid, thalf, nE);

  // --- Chebyshev recurrence: Tn = 2*S(Tc) - Tp ; acc += c_k*Tn ---
  float* Tp = A;
  float* Tc = Bm;
  for (int k = 2; k < K1; ++k) {
    applyS(Tc);
    cheby_step_kernel<<<gE, TB, 0, stream>>>(Tp, Tc, W, ACC, coeffs, k,
                                             tmid, thalf, nE);
    float* t = Tp; Tp = Tc; Tc = t;  // rotate (Tn was written into Tp's buffer)
  }

  // --- out[b, i] = acc[i, b] ---
  transpose_out_kernel<<<gT, bT, 0, stream>>>(ACC, (float*)d_out, ni);
}